// TrainableActivation_35424890257754
// MI455X (gfx1250) — compile-verified
//
#include <hip/hip_runtime.h>
#include <cstdint>
#include <cstddef>

// ---------------------------------------------------------------------------
// TrainableActivation (RBF activation), MI455X / gfx1250
//
// out[n,c,h,w] = sum_k w[c,k] * exp( -(x - mu_k)^2 / (2*sigma^2) )
//   mu_k = linspace(-1, 1, 31),  sigma = 2/30  =>  1/(2*sigma^2) = 112.5
//
// exp(-112.5*d^2) == exp2(C2 * d^2), C2 = -112.5 * log2(e) = -162.3031921
//
// Memory floor: 134 MB @ 23.3 TB/s ~ 5.8 us; kernel is v_exp_f32-bound,
// so per-bump op count is minimized (4 VALU + 1 trans), weights live in
// SGPRs (block-uniform channel), x streams through LDS via the CDNA5
// async-tensor path (global_load_async_to_lds_b128 + s_wait_asynccnt).
// ---------------------------------------------------------------------------

typedef float f32x4 __attribute__((ext_vector_type(4)));

#define NUM_W        31
#define NUM_CH       64
#define PLANE_ELEMS  16384            // 128*128, elements per (n,c) plane
#define BLOCK_T      256              // 8 waves (wave32)
#define TILES        4                // async tiles per block
#define ELEMS_PER_BLOCK (BLOCK_T * 4 * TILES)   // 4096 floats = 16 KB

// s_wait_asynccnt <= n  (ASYNCcnt tracks global_load_async_to_lds)
#define WAIT_ASYNC(n) asm volatile("s_wait_asynccnt " #n ::: "memory")

__device__ __forceinline__ void async_load_b128(uint32_t lds_off,
                                                const f32x4* gaddr) {
  // VDST = per-lane LDS byte address, VADDR = per-lane 64-bit global address
  asm volatile("global_load_async_to_lds_b128 %0, %1, off"
               :: "v"(lds_off), "v"(gaddr)
               : "memory");
}

__device__ __forceinline__ float rbf31(float xv, const float ws[NUM_W]) {
  const float C2 = -162.3031921f;     // -112.5 * log2(e)
  float acc = 0.0f;
#pragma unroll
  for (int k = 0; k < NUM_W; ++k) {
    const float mu = -1.0f + (float)k * (2.0f / 30.0f);  // folded per k
    float d  = xv - mu;
    float ar = d * (d * C2);
    acc = __builtin_fmaf(ws[k], __builtin_amdgcn_exp2f(ar), acc);
  }
  return acc;
}

extern "C" __global__ void __launch_bounds__(BLOCK_T)
rbf_act_kernel(const float* __restrict__ x, const float* __restrict__ w,
               float* __restrict__ out, int total) {
  __shared__ f32x4 lbuf[TILES][BLOCK_T];   // 16 KB

  const int    tid  = threadIdx.x;
  const size_t base = (size_t)blockIdx.x * ELEMS_PER_BLOCK;
  if (base >= (size_t)total) return;

  // 4 blocks per 16384-element plane -> channel is block-uniform,
  // so these 31 weight loads become scalar s_load into SGPRs.
  const int c = (int)((base / PLANE_ELEMS) % NUM_CH);
  float ws[NUM_W];
#pragma unroll
  for (int k = 0; k < NUM_W; ++k) ws[k] = w[(size_t)c * NUM_W + k];

  const f32x4* xg = (const f32x4*)(x + base);
  f32x4*       og = (f32x4*)(out + base);

  if (base + ELEMS_PER_BLOCK <= (size_t)total) {
    // ---- issue all async tiles (4 outstanding per wave) ----
#pragma unroll
    for (int t = 0; t < TILES; ++t) {
      uint32_t loff = (uint32_t)(uintptr_t)
          ((__attribute__((address_space(3))) f32x4*)&lbuf[t][tid]);
      async_load_b128(loff, xg + t * BLOCK_T + tid);
    }

    // Each lane consumes exactly the 16 B its own async load wrote,
    // so same-wave ASYNCcnt ordering is sufficient (no barriers).
    auto consume = [&](int t) {
      f32x4 v = lbuf[t][tid];
      f32x4 r;
      r.x = rbf31(v.x, ws);
      r.y = rbf31(v.y, ws);
      r.z = rbf31(v.z, ws);
      r.w = rbf31(v.w, ws);
      __builtin_nontemporal_store(r, og + t * BLOCK_T + tid);
    };

    WAIT_ASYNC(3); consume(0);
    WAIT_ASYNC(2); consume(1);
    WAIT_ASYNC(1); consume(2);
    WAIT_ASYNC(0); consume(3);
  } else {
    // generic tail (unused for the 16x64x128x128 shape, kept for safety)
    size_t rem = (size_t)total - base;
    for (size_t i = tid; i < rem; i += BLOCK_T) {
      out[base + i] = rbf31(x[base + i], ws);
    }
  }
}

// ---------------------------------------------------------------------------
extern "C" void kernel_launch(void* const* d_in, const int* in_sizes, int n_in,
                              void* d_out, int out_size, void* d_ws, size_t ws_size,
                              hipStream_t stream) {
  const float* x = (const float*)d_in[0];   // [16, 64, 128, 128] fp32
  const float* w = (const float*)d_in[1];   // [64, 31] fp32
  float* out     = (float*)d_out;           // same shape as x

  const int total   = in_sizes[0];
  const int nblocks = (total + ELEMS_PER_BLOCK - 1) / ELEMS_PER_BLOCK;

  rbf_act_kernel<<<nblocks, BLOCK_T, 0, stream>>>(x, w, out, total);
}